// DHCF_Encoder_35003983462553
// MI455X (gfx1250) — compile-verified
//
#include <hip/hip_runtime.h>
#include <hip/hip_bf16.h>
#include <stdint.h>

typedef __attribute__((ext_vector_type(2))) float v2f;
typedef __attribute__((ext_vector_type(8))) float v8f;

#define NEG_SLOPE 0.01f
#define P_KEEP    0.9f
#define EPSN      1e-12f

__device__ __forceinline__ uint32_t hash32(uint32_t x) {
  x ^= x >> 16; x *= 0x7feb352du;
  x ^= x >> 15; x *= 0x846ca68bu;
  x ^= x >> 16; return x;
}

// Copy layer-0 embedding (row-major, ld=64) into strided output (ld=stride).
__global__ void scatter_embed_kernel(const float* __restrict__ src,
                                     float* __restrict__ dst,
                                     int nrows, int stride) {
  int tid = blockIdx.x * blockDim.x + threadIdx.x;   // one float2 per thread
  int total = nrows * 32;
  if (tid >= total) return;
  int row = tid >> 5, c2 = tid & 31;
  v2f v = *(const v2f*)(src + (size_t)row * 64 + c2 * 2);
  *(v2f*)(dst + (size_t)row * stride + c2 * 2) = v;
}

// COO SpMM: acc[row] += val * X[col]; 32 lanes = one edge, 2 cols/lane.
// X and acc are L2-resident (25.6 MB each vs 192 MB L2); edge lists stream from HBM.
__global__ void spmm_kernel(const int* __restrict__ rows,
                            const int* __restrict__ cols,
                            const float* __restrict__ vals,
                            const float* __restrict__ X, int ldx,
                            float* __restrict__ acc, int nnz) {
  long long gid = (long long)blockIdx.x * blockDim.x + threadIdx.x;
  int e    = (int)(gid >> 5);
  int lane = (int)(gid & 31);
  if (e >= nnz) return;
  int r = rows[e], c = cols[e];
  float v = vals[e];
  v2f x = *(const v2f*)(X + (size_t)c * ldx + lane * 2);
  float* dst = acc + (size_t)r * 64 + lane * 2;
  atomicAdd(dst + 0, v * x.x);
  atomicAdd(dst + 1, v * x.y);
}

// Per 16-row tile: Y = acc@W + acc, LeakyReLU, dropout(hash), row L2-norm.
// One wave per tile; W staged in LDS in K-interleaved layout Ws[kc][n][4] so a
// lane's B-fragment (two consecutive K at one N) is a single aligned ds_load_b64.
__global__ void __launch_bounds__(256)
fused_gemm_epilogue(const float* __restrict__ acc, const float* __restrict__ W,
                    float* __restrict__ out, int stride, int colbase,
                    int ntiles, int nrows, uint32_t seed) {
  __shared__ float Ws[64 * 64];     // idx = kc*256 + n*4 + (k&3)
  for (int i = threadIdx.x; i < 64 * 64; i += blockDim.x) {
    int k = i >> 6, n = i & 63;     // W is row-major [k][n]
    Ws[(k >> 2) * 256 + n * 4 + (k & 3)] = W[i];
  }
  __syncthreads();

  int wave = threadIdx.x >> 5;
  int lane = threadIdx.x & 31;
  int tile = blockIdx.x * (blockDim.x >> 5) + wave;
  if (tile >= ntiles) return;          // uniform per wave: EXEC all-1s for WMMA

  int r0   = tile * 16;
  int m    = lane & 15;                // 0..15
  int koff = (lane >> 4) * 2;          // lanes 16-31 hold K+2,K+3 of A/B frags

  int arow = r0 + m; if (arow >= nrows) arow = nrows - 1;
  const float* aptr = acc + (size_t)arow * 64;

  v8f c0 = {}, c1 = {}, c2 = {}, c3 = {};
#pragma unroll
  for (int kc = 0; kc < 16; ++kc) {
    v2f a = *(const v2f*)(aptr + kc * 4 + koff);     // A[M][kb], A[M][kb+1]
    const float* wbase = Ws + kc * 256 + koff;       // + n*4
    v2f b0 = *(const v2f*)(wbase + ( 0 + m) * 4);    // one ds_load_b64 each,
    v2f b1 = *(const v2f*)(wbase + (16 + m) * 4);    // bank-conflict-free
    v2f b2 = *(const v2f*)(wbase + (32 + m) * 4);
    v2f b3 = *(const v2f*)(wbase + (48 + m) * 4);
    c0 = __builtin_amdgcn_wmma_f32_16x16x4_f32(false, a, false, b0, (short)0, c0, false, false);
    c1 = __builtin_amdgcn_wmma_f32_16x16x4_f32(false, a, false, b1, (short)0, c1, false, false);
    c2 = __builtin_amdgcn_wmma_f32_16x16x4_f32(false, a, false, b2, (short)0, c2, false, false);
    c3 = __builtin_amdgcn_wmma_f32_16x16x4_f32(false, a, false, b3, (short)0, c3, false, false);
  }

  // C layout: VGPR r, lanes 0-15 -> M=r, lanes 16-31 -> M=r+8; N = nt*16 + m.
  int halfsel = lane >> 4;
  float x[4][8];
  float ss[8];
#pragma unroll
  for (int r = 0; r < 8; ++r) ss[r] = 0.f;
#pragma unroll
  for (int r = 0; r < 8; ++r) {
    int row = r0 + r + 8 * halfsel;
    int rl  = row < nrows ? row : nrows - 1;
#pragma unroll
    for (int nt = 0; nt < 4; ++nt) {
      int col  = nt * 16 + m;
      float cv = (nt == 0 ? c0[r] : nt == 1 ? c1[r] : nt == 2 ? c2[r] : c3[r]);
      float v  = cv + acc[(size_t)rl * 64 + col];          // residual (+nu)
      v = v >= 0.f ? v : NEG_SLOPE * v;                    // LeakyReLU
      uint32_t h = hash32(seed ^ (uint32_t)(rl * 64 + col));
      v *= ((h * 2.3283064365386963e-10f) < P_KEEP) ? (1.0f / P_KEEP) : 0.0f;
      x[nt][r] = v;
      ss[r] += v * v;
    }
  }
  // Row sum-of-squares: reduce across the 16 lanes of each half-wave (wave32).
#pragma unroll
  for (int off = 1; off < 16; off <<= 1) {
#pragma unroll
    for (int r = 0; r < 8; ++r) ss[r] += __shfl_xor(ss[r], off, 32);
  }
#pragma unroll
  for (int r = 0; r < 8; ++r) {
    int row = r0 + r + 8 * halfsel;
    if (row >= nrows) continue;
    float inv = 1.0f / fmaxf(sqrtf(ss[r]), EPSN);
#pragma unroll
    for (int nt = 0; nt < 4; ++nt)
      out[(size_t)row * stride + colbase + nt * 16 + m] = x[nt][r] * inv;
  }
}

extern "C" void kernel_launch(void* const* d_in, const int* in_sizes, int n_in,
                              void* d_out, int out_size, void* d_ws, size_t ws_size,
                              hipStream_t stream) {
  const float* user_emb = (const float*)d_in[0];
  const float* item_emb = (const float*)d_in[1];
  const float* weights  = (const float*)d_in[2];
  const float* hu_vals  = (const float*)d_in[3];
  const float* hi_vals  = (const float*)d_in[4];
  const int*   hu_rows  = (const int*)d_in[5];
  const int*   hu_cols  = (const int*)d_in[6];
  const int*   hi_rows  = (const int*)d_in[7];
  const int*   hi_cols  = (const int*)d_in[8];

  const int NU  = in_sizes[0] / 64;
  const int NI  = in_sizes[1] / 64;
  const int NL  = in_sizes[2] / (64 * 64);
  const int NNZ = in_sizes[3];
  const int stride = (NL + 1) * 64;          // 192

  float* outU = (float*)d_out;
  float* outI = outU + (size_t)NU * stride;
  float* accU = (float*)d_ws;
  float* accI = accU + (size_t)NU * 64;

  // Layer-0 columns of the concat output.
  scatter_embed_kernel<<<(NU * 32 + 255) / 256, 256, 0, stream>>>(user_emb, outU, NU, stride);
  scatter_embed_kernel<<<(NI * 32 + 255) / 256, 256, 0, stream>>>(item_emb, outI, NI, stride);

  const long long spmmThreads = (long long)NNZ * 32;
  const int spmmBlocks = (int)((spmmThreads + 255) / 256);
  const int tilesU = (NU + 15) / 16, tilesI = (NI + 15) / 16;

  for (int k = 0; k < NL; ++k) {
    hipMemsetAsync(d_ws, 0, (size_t)(NU + NI) * 64 * sizeof(float), stream);

    spmm_kernel<<<spmmBlocks, 256, 0, stream>>>(hu_rows, hu_cols, hu_vals,
                                                outU + (size_t)64 * k, stride, accU, NNZ);
    spmm_kernel<<<spmmBlocks, 256, 0, stream>>>(hi_rows, hi_cols, hi_vals,
                                                outI + (size_t)64 * k, stride, accI, NNZ);

    const float* W = weights + (size_t)k * 64 * 64;
    fused_gemm_epilogue<<<(tilesU + 7) / 8, 256, 0, stream>>>(
        accU, W, outU, stride, 64 * (k + 1), tilesU, NU,
        0x9E3779B9u * (uint32_t)(2 * k + 1));
    fused_gemm_epilogue<<<(tilesI + 7) / 8, 256, 0, stream>>>(
        accI, W, outI, stride, 64 * (k + 1), tilesI, NI,
        0x9E3779B9u * (uint32_t)(2 * k + 2));
  }
}